// GCNModel_24764781429205
// MI455X (gfx1250) — compile-verified
//
#include <hip/hip_runtime.h>

typedef float v2f __attribute__((ext_vector_type(2)));
typedef float v4f __attribute__((ext_vector_type(4)));
typedef float v8f __attribute__((ext_vector_type(8)));

static __device__ __forceinline__ v8f wmma4(v2f a, v2f b, v8f c) {
  return __builtin_amdgcn_wmma_f32_16x16x4_f32(false, a, false, b, (short)0, c,
                                               false, false);
}

// ---------------- degree / normalization ----------------

__global__ void k_init_deg(float* deg, int N) {
  int i = blockIdx.x * blockDim.x + threadIdx.x;
  if (i < N) deg[i] = 1.0f;  // self-loop weight 1
}

__global__ void k_deg_scatter(const long long* __restrict__ dst,
                              const float* __restrict__ w,
                              float* deg, int E) {
  int e = blockIdx.x * blockDim.x + threadIdx.x;
  if (e < E) unsafeAtomicAdd(&deg[dst[e]], w[e]);
}

__global__ void k_deg_to_dis(float* deg, int N) {
  int i = blockIdx.x * blockDim.x + threadIdx.x;
  if (i < N) {
    float d = deg[i];
    deg[i] = d > 0.0f ? rsqrtf(fmaxf(d, 1e-12f)) : 0.0f;
  }
}

__global__ void k_norm(const long long* __restrict__ src,
                       const long long* __restrict__ dst,
                       const float* __restrict__ w,
                       const float* __restrict__ dis,
                       float* __restrict__ norm, int E) {
  int e = blockIdx.x * blockDim.x + threadIdx.x;
  if (e < E) norm[e] = dis[src[e]] * w[e] * dis[dst[e]];
}

// ---------------- GEMM 1: h1 = x[M,4] @ W1[4,128] ----------------
// One wave computes a 16x128 stripe: 1 A-fragment load, 8 WMMAs.

__global__ void __launch_bounds__(256) k_gemm1(const float* __restrict__ x,
                                               const float* __restrict__ W1,
                                               float* __restrict__ h1, int M) {
  const int mTile  = (int)(blockIdx.x * 8 + (threadIdx.x >> 5));
  const int mTiles = (M + 15) >> 4;
  if (mTile >= mTiles) return;  // wave-uniform
  const int lane = threadIdx.x & 31;
  const int half = lane >> 4, r = lane & 15;
  const int m0 = mTile << 4;
  int mrow = m0 + r; if (mrow > M - 1) mrow = M - 1;

  // A 16x4 f32: lanes 0-15 hold K=0,1 ; lanes 16-31 hold K=2,3 (row = lane%16)
  v2f a = *(const v2f*)(x + (size_t)mrow * 4 + half * 2);

  v8f acc[8];
#pragma unroll
  for (int nt = 0; nt < 8; ++nt) {
    v2f b;  // B 4x16: lane holds column nt*16+r, rows K = half*2 + {0,1}
    b.x = W1[(half * 2 + 0) * 128 + nt * 16 + r];
    b.y = W1[(half * 2 + 1) * 128 + nt * 16 + r];
    v8f z = {0.f, 0.f, 0.f, 0.f, 0.f, 0.f, 0.f, 0.f};
    acc[nt] = wmma4(a, b, z);
  }

  float* p = h1 + (size_t)(m0 + half * 8) * 128 + r;
  if (m0 + 16 <= M) {  // full tile: unguarded immediate-offset store clause
#pragma unroll
    for (int nt = 0; nt < 8; ++nt)
#pragma unroll
      for (int i = 0; i < 8; ++i)
        p[(size_t)i * 128 + nt * 16] = acc[nt][i];
  } else {
    for (int nt = 0; nt < 8; ++nt)
      for (int i = 0; i < 8; ++i) {
        int m = m0 + half * 8 + i;
        if (m < M) h1[(size_t)m * 128 + nt * 16 + r] = acc[nt][i];
      }
  }
}

// ---------------- GEMM 2: h2 = relu(agg1[M,128] + b1) @ W2[128,64] ----------------
// W2 staged transposed in LDS (stride 132 => conflict-free b64 reads).
// One wave computes a 16x64 stripe: A fragment loaded once per K-step, 4 WMMAs.

__global__ void __launch_bounds__(256) k_gemm2(const float* __restrict__ agg1,
                                               const float* __restrict__ b1,
                                               const float* __restrict__ W2,
                                               float* __restrict__ h2, int M) {
  __shared__ float sWT[64 * 132];  // sWT[n*132 + k] = W2[k*64 + n]
  __shared__ float sB[128];
  const int tid = threadIdx.x;
#pragma unroll
  for (int i = 0; i < 32; ++i) {  // 8192 elements / 256 threads
    int idx = i * 256 + tid;
    int k = idx >> 6, n = idx & 63;
    sWT[n * 132 + k] = W2[idx];
  }
  if (tid < 128) sB[tid] = b1[tid];
  __syncthreads();

  const int mTile  = (int)(blockIdx.x * 8 + (tid >> 5));
  const int mTiles = (M + 15) >> 4;
  if (mTile >= mTiles) return;  // wave-uniform, after the barrier
  const int lane = tid & 31;
  const int half = lane >> 4, r = lane & 15;
  const int m0 = mTile << 4;
  int mrow = m0 + r; if (mrow > M - 1) mrow = M - 1;
  const float* aRow = agg1 + (size_t)mrow * 128;

  v8f acc[4];
#pragma unroll
  for (int nt = 0; nt < 4; ++nt) {
    v8f z = {0.f, 0.f, 0.f, 0.f, 0.f, 0.f, 0.f, 0.f};
    acc[nt] = z;
  }

#pragma unroll 4
  for (int k0 = 0; k0 < 128; k0 += 4) {
    const int ka = k0 + half * 2;
    v2f av = *(const v2f*)(aRow + ka);
    v2f bi = *(const v2f*)(sB + ka);
    v2f a;
    a.x = fmaxf(av.x + bi.x, 0.0f);  // fused bias + ReLU of layer 1
    a.y = fmaxf(av.y + bi.y, 0.0f);
#pragma unroll
    for (int nt = 0; nt < 4; ++nt) {
      v2f b = *(const v2f*)(sWT + (nt * 16 + r) * 132 + ka);
      acc[nt] = wmma4(a, b, acc[nt]);
    }
  }

  float* p = h2 + (size_t)(m0 + half * 8) * 64 + r;
  if (m0 + 16 <= M) {
#pragma unroll
    for (int nt = 0; nt < 4; ++nt)
#pragma unroll
      for (int i = 0; i < 8; ++i)
        p[(size_t)i * 64 + nt * 16] = acc[nt][i];
  } else {
    for (int nt = 0; nt < 4; ++nt)
      for (int i = 0; i < 8; ++i) {
        int m = m0 + half * 8 + i;
        if (m < M) h2[(size_t)m * 64 + nt * 16 + r] = acc[nt][i];
      }
  }
}

// ---------------- self-loop init + edge scatter (per layer) ----------------

__global__ void k_agg_init4(const float* __restrict__ h, const float* __restrict__ dis,
                            float* __restrict__ agg, int logF, int total4) {
  int t = blockIdx.x * blockDim.x + threadIdx.x;
  if (t >= total4) return;
  float d = dis[t >> (logF - 2)];
  d *= d;
  v4f v = *(const v4f*)(h + (size_t)t * 4);
  v4f o;
  o.x = d * v.x; o.y = d * v.y; o.z = d * v.z; o.w = d * v.w;
  *(v4f*)(agg + (size_t)t * 4) = o;
}

// wave per edge: src/dst/norm fetched once, per-lane vector gather, f32 atomics
template <int VEC>
__global__ void __launch_bounds__(256) k_agg_scatter_w(
    const float* __restrict__ h, const long long* __restrict__ src,
    const long long* __restrict__ dst, const float* __restrict__ norm,
    float* __restrict__ agg, int E) {
  const int e = (int)(blockIdx.x * 8 + (threadIdx.x >> 5));
  if (e >= E) return;
  const int lane = threadIdx.x & 31;
  const int F = VEC * 32;
  const int s = (int)src[e];
  const int d = (int)dst[e];
  const float nw = norm[e];
  const float* hp = h + (size_t)s * F + lane * VEC;
  float* ap = agg + (size_t)d * F + lane * VEC;
  if (VEC == 4) {
    v4f v = *(const v4f*)hp;
    unsafeAtomicAdd(ap + 0, nw * v.x);
    unsafeAtomicAdd(ap + 1, nw * v.y);
    unsafeAtomicAdd(ap + 2, nw * v.z);
    unsafeAtomicAdd(ap + 3, nw * v.w);
  } else {
    v2f v = *(const v2f*)hp;
    unsafeAtomicAdd(ap + 0, nw * v.x);
    unsafeAtomicAdd(ap + 1, nw * v.y);
  }
}

// ---------------- layer 3: h3[i] = relu(agg2[i,:]+b2) . W3  (wave per node) ----------------

__global__ void __launch_bounds__(256) k_layer3(const float* __restrict__ agg2,
                                                const float* __restrict__ b2,
                                                const float* __restrict__ W3,
                                                float* __restrict__ h3, int N) {
  int node = (int)(blockIdx.x * 8 + (threadIdx.x >> 5));
  if (node >= N) return;
  int lane = threadIdx.x & 31;
  v2f v  = *(const v2f*)(agg2 + (size_t)node * 64 + lane * 2);
  v2f bb = *(const v2f*)(b2 + lane * 2);
  v2f ww = *(const v2f*)(W3 + lane * 2);
  float s = fmaxf(v.x + bb.x, 0.0f) * ww.x + fmaxf(v.y + bb.y, 0.0f) * ww.y;
#pragma unroll
  for (int off = 16; off > 0; off >>= 1) s += __shfl_xor(s, off, 32);
  if (lane == 0) h3[node] = s;
}

__global__ void k_out_init(const float* __restrict__ h3, const float* __restrict__ dis,
                           const float* __restrict__ b3, float* __restrict__ out, int N) {
  int i = blockIdx.x * blockDim.x + threadIdx.x;
  if (i < N) {
    float d = dis[i];
    out[i] = b3[0] + d * d * h3[i];
  }
}

__global__ void k_out_scatter(const float* __restrict__ h3,
                              const long long* __restrict__ src,
                              const long long* __restrict__ dst,
                              const float* __restrict__ norm,
                              float* out, int E) {
  int e = blockIdx.x * blockDim.x + threadIdx.x;
  if (e < E) unsafeAtomicAdd(&out[dst[e]], norm[e] * h3[src[e]]);
}

// ---------------- host ----------------

extern "C" void kernel_launch(void* const* d_in, const int* in_sizes, int n_in,
                              void* d_out, int out_size, void* d_ws, size_t ws_size,
                              hipStream_t stream) {
  const float*     x  = (const float*)d_in[0];
  const long long* ei = (const long long*)d_in[1];  // int64 [2, E]
  const float*     w  = (const float*)d_in[2];
  const float*     W1 = (const float*)d_in[3];
  const float*     b1 = (const float*)d_in[4];
  const float*     W2 = (const float*)d_in[5];
  const float*     b2 = (const float*)d_in[6];
  const float*     W3 = (const float*)d_in[7];
  const float*     b3 = (const float*)d_in[8];
  float* out = (float*)d_out;

  const int N = in_sizes[0] / 4;  // 100000
  const int E = in_sizes[2];      // 800000
  const long long* src = ei;
  const long long* dst = ei + E;

  // workspace layout: dis[N] | norm[E] | bufA[N*128] | bufB[N*128]  (~106 MB)
  char* ws = (char*)d_ws;
  size_t off = 0;
  auto align256 = [](size_t v) { return (v + 255) & ~(size_t)255; };
  float* dis  = (float*)(ws + off); off = align256(off + (size_t)N * 4);
  float* norm = (float*)(ws + off); off = align256(off + (size_t)E * 4);
  float* bufA = (float*)(ws + off); off = align256(off + (size_t)N * 128 * 4);
  float* bufB = (float*)(ws + off); off = align256(off + (size_t)N * 128 * 4);
  (void)ws_size; (void)n_in; (void)out_size;

  const int B = 256;
  dim3 blk(B);
  auto grid1 = [&](long long t) { return dim3((unsigned)((t + B - 1) / B)); };
  auto gridW = [&](long long waves) {  // 8 waves per 256-thread block
    return dim3((unsigned)((waves + 7) / 8));
  };

  // normalization coefficients
  k_init_deg   <<<grid1(N), blk, 0, stream>>>(dis, N);
  k_deg_scatter<<<grid1(E), blk, 0, stream>>>(dst, w, dis, E);
  k_deg_to_dis <<<grid1(N), blk, 0, stream>>>(dis, N);
  k_norm       <<<grid1(E), blk, 0, stream>>>(src, dst, w, dis, norm, E);

  const long long mTiles = (N + 15) / 16;

  // layer 1: transform (WMMA) -> self-loop init -> edge scatter
  k_gemm1           <<<gridW(mTiles), blk, 0, stream>>>(x, W1, bufA, N);
  k_agg_init4       <<<grid1((long long)N * 32), blk, 0, stream>>>(bufA, dis, bufB, 7, N * 32);
  k_agg_scatter_w<4><<<gridW(E), blk, 0, stream>>>(bufA, src, dst, norm, bufB, E);

  // layer 2: relu(agg1+b1) fused into WMMA GEMM (W2 staged in LDS)
  k_gemm2           <<<gridW(mTiles), blk, 0, stream>>>(bufB, b1, W2, bufA, N);
  k_agg_init4       <<<grid1((long long)N * 16), blk, 0, stream>>>(bufA, dis, bufB, 6, N * 16);
  k_agg_scatter_w<2><<<gridW(E), blk, 0, stream>>>(bufA, src, dst, norm, bufB, E);

  // layer 3: relu(agg2+b2) . W3, then aggregate directly into d_out
  k_layer3     <<<gridW(N), blk, 0, stream>>>(bufB, b2, W3, bufA, N);
  k_out_init   <<<grid1(N), blk, 0, stream>>>(bufA, dis, b3, out, N);
  k_out_scatter<<<grid1(E), blk, 0, stream>>>(bufA, src, dst, norm, out, E);
}